// HyperbolicLayeredLoss_75136157876229
// MI455X (gfx1250) — compile-verified
//
#include <hip/hip_runtime.h>
#include <cstdint>
#include <cstddef>

// ---------------------------------------------------------------------------
// HyperbolicLayeredLoss for MI455X (gfx1250, wave32, WMMA)
//   Pass 1 (norms)      : row squared-norms
//   Pass 2 (partial)    : fused Gram (WMMA f32 16x16x4) -> z-proxy -> partial
//                         per-row top-5. Column sweep split 8 ways (4096
//                         waves). B tiles streamed global->LDS with the
//                         gfx1250 async copy engine (double-buffered,
//                         ASYNCcnt-pipelined); WMMA operands fed by
//                         bank-conflict-free ds_load_b64.
//   Pass 3 (merge)      : merge 8 partial top-5 lists/row, arccosh+logsumexp,
//                         deterministic PCG negative sampling
//   Pass 4 (reduce)     : deterministic tree reduction -> scalar
// ---------------------------------------------------------------------------

#define B_N        4096
#define D_K        64
#define TOPK       5
#define NEG_S      20
#define NSPLIT     8            // column-sweep splits per row tile
#define CT_PER     (B_N / 16 / NSPLIT)   // 32 column tiles per wave
#define TAU_F      0.1f
#define LAMBDA_H_F 0.01f
#define EPS_F      1e-9f
#define HUGE_D     3.0e38f
#define COL_STRIDE 68           // floats per column in LDS (64 + 4 pad)

typedef __attribute__((ext_vector_type(2))) float v2f;
typedef __attribute__((ext_vector_type(8))) float v8f;

__device__ __forceinline__ uint32_t pcg_hash(uint32_t x) {
    x = x * 747796405u + 2891336453u;
    uint32_t w = ((x >> ((x >> 28u) + 4u)) ^ x) * 277803737u;
    return (w >> 22u) ^ w;
}

__device__ __forceinline__ float acosh_pos(float z) {
    // z >= 1+eps guaranteed by construction
    return __logf(z + __fsqrt_rn(z * z - 1.0f));
}

// keep 5 smallest, ascending; constant indices only (stays in VGPRs)
__device__ __forceinline__ void topk_insert(float d, int idx,
                                            float (&v)[TOPK], int (&ix)[TOPK]) {
    if (d < v[4]) {
        v[4] = d; ix[4] = idx;
        if (v[4] < v[3]) { float t=v[4]; v[4]=v[3]; v[3]=t; int u=ix[4]; ix[4]=ix[3]; ix[3]=u; }
        if (v[3] < v[2]) { float t=v[3]; v[3]=v[2]; v[2]=t; int u=ix[3]; ix[3]=ix[2]; ix[2]=u; }
        if (v[2] < v[1]) { float t=v[2]; v[2]=v[1]; v[1]=t; int u=ix[2]; ix[2]=ix[1]; ix[1]=u; }
        if (v[1] < v[0]) { float t=v[1]; v[1]=v[0]; v[0]=t; int u=ix[1]; ix[1]=ix[0]; ix[0]=u; }
    }
}

// --------------------------- pass 1: ||x_i||^2 -----------------------------
__global__ __launch_bounds__(256) void hll_norms_kernel(
        const float* __restrict__ audio, const float* __restrict__ text,
        float* __restrict__ norms) {
    int i = blockIdx.x * blockDim.x + threadIdx.x;     // 0 .. 2*B_N-1
    if (i >= 2 * B_N) return;
    const float* X = (i < B_N) ? audio : text;
    int r = (i < B_N) ? i : i - B_N;
    const float4* p = (const float4*)(X + (size_t)r * D_K);
    float s = 0.0f;
#pragma unroll
    for (int k = 0; k < D_K / 4; ++k) {
        float4 v = p[k];
        s += v.x * v.x + v.y * v.y + v.z * v.z + v.w * v.w;
    }
    norms[i] = s;
}

// ---------------- pass 2: WMMA Gram + z-proxy + partial top-5 --------------
// grid: (B_N/16, NSPLIT, 2), one wave32 per block. Wave owns 16 rows and
// sweeps CT_PER column tiles of its split.
__global__ __launch_bounds__(32) void hll_partial_topk_kernel(
        const float* __restrict__ audio, const float* __restrict__ text,
        const float* __restrict__ norms,
        float* __restrict__ ws_z, int* __restrict__ ws_i) {
    const int lane = threadIdx.x;          // 0..31
    const int half = lane >> 4;            // 0/1
    const int l16  = lane & 15;
    const int rowTile = blockIdx.x;        // 0..255
    const int cs      = blockIdx.y;        // 0..NSPLIT-1
    const int mod     = blockIdx.z;        // 0: audio, 1: text

    const float* X   = (mod == 0) ? audio : text;
    const float* nrm = norms + (size_t)mod * B_N;
    const int rowBase = rowTile * 16;

    __shared__ float bbuf[2][16 * COL_STRIDE];  // double-buffered B tiles
    __shared__ float tile[16 * 17];             // padded transpose buffer
    __shared__ float mv[16 * TOPK];
    __shared__ int   mi[16 * TOPK];

    // ---- A fragments: rows rowBase..+15, full K=64, resident in VGPRs ----
    // f32 16x4 A layout: lanes 0-15 -> M=lane, VGPR0 K={0|2}, VGPR1 K={1|3}
    const int rA = rowBase + l16;
    v2f afrag[16];
    {
        const float* ap = X + (size_t)rA * D_K + 2 * half;
#pragma unroll
        for (int kk = 0; kk < 16; ++kk)
            afrag[kk] = *(const v2f*)(ap + 4 * kk);
    }
    // per-lane row norms for this lane's 8 C rows (M = j + 8*half);
    // rrc = 1/(1-clamp(xs)) hoisted: removes per-element IEEE divide
    float rn[8], rrc[8];
#pragma unroll
    for (int j = 0; j < 8; ++j) {
        float xs = nrm[rowBase + j + 8 * half];
        rn[j]  = xs;
        rrc[j] = __builtin_amdgcn_rcpf(1.0f - fminf(xs, 1.0f - EPS_F));
    }

    // per-lane top-5 over z-proxy (lane r: cols 0-7 of tiles; lane r+16: 8-15)
    float t5v[TOPK]; int t5i[TOPK];
#pragma unroll
    for (int j = 0; j < TOPK; ++j) { t5v[j] = HUGE_D; t5i[j] = -1; }

    // --------------- async global->LDS B-tile copy (8 x b128/lane) --------
    // 4KB tile = 256 16B chunks; lane handles chunks lane+32*i.
    // LDS layout: column c at c*COL_STRIDE floats (stride 272B) so the
    // ds_load_b64 feeds below hit banks 4*l16 mod 64: conflict-free.
    const uint32_t lds0 = (uint32_t)(uintptr_t)&bbuf[0][0];
    auto issue_tile_copy = [&](int ct, int stage) {
        const float* gbase = X + (size_t)ct * 16 * D_K;
        const uint32_t sbase = lds0 + (uint32_t)stage * (16 * COL_STRIDE * 4);
#pragma unroll
        for (int i = 0; i < 8; ++i) {
            const int chunk = lane + 32 * i;
            const int col   = chunk >> 4;
            const int piece = chunk & 15;
            const uint32_t laddr = sbase + (uint32_t)(col * (COL_STRIDE * 4) + piece * 16);
            const void* g = (const void*)(gbase + chunk * 4);
            asm volatile("global_load_async_to_lds_b128 %0, %1, off"
                         :: "v"(laddr), "v"(g) : "memory");
        }
    };

    const int ct0 = cs * CT_PER;
    issue_tile_copy(ct0, 0);                       // prologue: stage 0

    // ------------------------- column-tile sweep --------------------------
    for (int t = 0; t < CT_PER; ++t) {
        const int ct   = ct0 + t;
        const int colB = ct * 16 + l16;

        // stream next tile into the other stage (dummy refill on last iter)
        const int ctn = (t + 1 < CT_PER) ? (ct + 1) : ct0;
        issue_tile_copy(ctn, (t + 1) & 1);
        // in-order completion: <=8 outstanding => current stage resident
        asm volatile("s_wait_asynccnt 0x8" ::: "memory");

        // B fragments from LDS (conflict-free ds_load_b64)
        const float* bsrc = &bbuf[t & 1][l16 * COL_STRIDE + 2 * half];
        v2f bfrag[16];
#pragma unroll
        for (int kk = 0; kk < 16; ++kk)
            bfrag[kk] = *(const v2f*)(bsrc + 4 * kk);
        const float ys = nrm[colB];

        v8f c8 = {};
#pragma unroll
        for (int kk = 0; kk < 16; ++kk)
            c8 = __builtin_amdgcn_wmma_f32_16x16x4_f32(
                    false, afrag[kk], false, bfrag[kk], (short)0, c8,
                    false, false);

        // epilogue: C entry (M = j+8*half, N = l16) -> z proxy -> LDS
        const float ryc  = __builtin_amdgcn_rcpf(1.0f - fminf(ys, 1.0f - EPS_F));
        const float ryc2 = 2.0f * ryc;
#pragma unroll
        for (int j = 0; j < 8; ++j) {
            const int gm = rowBase + j + 8 * half;
            float diff = fmaxf(rn[j] + ys - 2.0f * c8[j], 0.0f);
            float z    = fmaxf(fmaf(diff * rrc[j], ryc2, 1.0f), 1.0f + EPS_F);
            if (gm == colB) z = HUGE_D;                 // diagonal -> inf
            tile[(j + 8 * half) * 17 + l16] = z;
        }
        __syncthreads();
        // transpose-scan: lane r owns row r cols 0-7, lane r+16 cols 8-15
        const int cbase = ct * 16 + 8 * half;
#pragma unroll
        for (int c = 0; c < 8; ++c) {
            float z = tile[l16 * 17 + 8 * half + c];
            topk_insert(z, cbase + c, t5v, t5i);
        }
        __syncthreads();
    }

    // ------------- merge half-lane lists, store partial top-5 -------------
    if (half) {
#pragma unroll
        for (int j = 0; j < TOPK; ++j) {
            mv[l16 * TOPK + j] = t5v[j];
            mi[l16 * TOPK + j] = t5i[j];
        }
    }
    __syncthreads();
    if (!half) {
#pragma unroll
        for (int j = 0; j < TOPK; ++j)
            topk_insert(mv[l16 * TOPK + j], mi[l16 * TOPK + j], t5v, t5i);
        const int R = rowBase + l16;
        const size_t base =
            (((size_t)mod * B_N + R) * NSPLIT + cs) * TOPK;
#pragma unroll
        for (int j = 0; j < TOPK; ++j) {
            ws_z[base + j] = t5v[j];
            ws_i[base + j] = t5i[j];
        }
    }
    // trailing dummy async copy is drained by S_ENDPGM's implicit wait-idle
}

// ------------- pass 3: merge splits + logsumexp + negatives ----------------
// one thread per (modality, row): 8192 threads
__global__ __launch_bounds__(256) void hll_merge_loss_kernel(
        const float* __restrict__ audio, const float* __restrict__ text,
        const float* __restrict__ norms,
        const float* __restrict__ ws_z, const int* __restrict__ ws_i,
        float* __restrict__ rowloss) {
    int i = blockIdx.x * blockDim.x + threadIdx.x;
    if (i >= 2 * B_N) return;
    const int mod = i >> 12;               // /4096
    const int R   = i & (B_N - 1);
    const float* X   = (mod == 0) ? audio : text;
    const float* nrm = norms + (size_t)mod * B_N;

    // merge the NSPLIT partial top-5 lists (z-proxy domain)
    float t5v[TOPK]; int t5i[TOPK];
#pragma unroll
    for (int j = 0; j < TOPK; ++j) { t5v[j] = HUGE_D; t5i[j] = -1; }
    const size_t base = ((size_t)mod * B_N + R) * NSPLIT * TOPK;
#pragma unroll
    for (int s = 0; s < NSPLIT; ++s)
#pragma unroll
        for (int j = 0; j < TOPK; ++j)
            topk_insert(ws_z[base + s * TOPK + j],
                        ws_i[base + s * TOPK + j], t5v, t5i);

    // pos_term = logsumexp(-d/tau) over top-5 (arccosh only here)
    float pd[TOPK];
#pragma unroll
    for (int j = 0; j < TOPK; ++j) pd[j] = acosh_pos(t5v[j]);
    float mp = -pd[0] / TAU_F;             // smallest d -> max of -d/tau
    float sp = 0.0f;
#pragma unroll
    for (int j = 0; j < TOPK; ++j) sp += __expf(-pd[j] / TAU_F - mp);
    float pos_term = mp + __logf(sp);

    // sort top-5 indices ascending (Bose-Nelson network, const indices)
    int si[TOPK];
#pragma unroll
    for (int j = 0; j < TOPK; ++j) si[j] = t5i[j];
#define CSW(a,b) { if (si[a] > si[b]) { int t=si[a]; si[a]=si[b]; si[b]=t; } }
    CSW(0,1) CSW(3,4) CSW(2,4) CSW(2,3) CSW(0,3) CSW(0,2) CSW(1,4) CSW(1,3) CSW(1,2)
#undef CSW

    // 20 deterministic negatives from the (B_N - 5)-column pool
    const float xs  = nrm[R];
    const float rxc = __builtin_amdgcn_rcpf(1.0f - fminf(xs, 1.0f - EPS_F));
    const uint32_t seedbase =
        (mod ? 0x9E3779B9u : 0x85EBCA6Bu) ^ (uint32_t)(R * NEG_S);
    float nd[NEG_S];
#pragma unroll
    for (int s = 0; s < NEG_S; ++s) {
        uint32_t h = pcg_hash(seedbase + (uint32_t)s);
        int col = (int)(h % (uint32_t)(B_N - TOPK));
#pragma unroll
        for (int j = 0; j < TOPK; ++j)
            if (si[j] <= col) col++;       // map pool idx -> column
        if (col == R) {
            nd[s] = HUGE_D;                // diagonal: exp -> 0
        } else {
            const float4* xr = (const float4*)(X + (size_t)R   * D_K);
            const float4* xc = (const float4*)(X + (size_t)col * D_K);
            float dot = 0.0f;
#pragma unroll
            for (int k = 0; k < D_K / 4; ++k) {
                float4 a = xr[k], b = xc[k];
                dot += a.x * b.x + a.y * b.y + a.z * b.z + a.w * b.w;
            }
            float ys  = nrm[col];
            float ryc = __builtin_amdgcn_rcpf(1.0f - fminf(ys, 1.0f - EPS_F));
            float diff = fmaxf(xs + ys - 2.0f * dot, 0.0f);
            float z = fmaxf(fmaf(diff * rxc, 2.0f * ryc, 1.0f), 1.0f + EPS_F);
            nd[s] = acosh_pos(z);
        }
    }
    float mn = -3.4e38f;
#pragma unroll
    for (int s = 0; s < NEG_S; ++s) mn = fmaxf(mn, -nd[s] / TAU_F);
    float sn = 0.0f;
#pragma unroll
    for (int s = 0; s < NEG_S; ++s) sn += __expf(-nd[s] / TAU_F - mn);
    float neg_term = mn + __logf(sn);

    rowloss[(size_t)mod * B_N + R] = neg_term - pos_term;
}

// -------------------- pass 4: deterministic reduction ----------------------
__global__ __launch_bounds__(256) void hll_reduce_kernel(
        const float* __restrict__ rowloss, float* __restrict__ out) {
    __shared__ float sm[256];
    int t = threadIdx.x;
    float s = 0.0f;
    for (int i = t; i < 2 * B_N; i += 256) s += rowloss[i];
    sm[t] = s;
    __syncthreads();
    for (int off = 128; off > 0; off >>= 1) {
        if (t < off) sm[t] += sm[t + off];
        __syncthreads();
    }
    if (t == 0) out[0] = LAMBDA_H_F * sm[0] / (float)B_N;
}

// ---------------------------------------------------------------------------
extern "C" void kernel_launch(void* const* d_in, const int* in_sizes, int n_in,
                              void* d_out, int out_size, void* d_ws, size_t ws_size,
                              hipStream_t stream) {
    (void)in_sizes; (void)n_in; (void)out_size; (void)ws_size;
    const float* audio = (const float*)d_in[0];   // [4096,64] f32
    const float* text  = (const float*)d_in[1];   // [4096,64] f32

    float* ws      = (float*)d_ws;
    float* norms   = ws;                                   // 8192 f32
    float* rowloss = norms + 2 * B_N;                      // 8192 f32
    float* ws_z    = rowloss + 2 * B_N;                    // 2*4096*8*5 f32
    int*   ws_i    = (int*)(ws_z + (size_t)2 * B_N * NSPLIT * TOPK);

    hll_norms_kernel<<<(2 * B_N + 255) / 256, 256, 0, stream>>>(audio, text, norms);

    dim3 grid(B_N / 16, NSPLIT, 2);                        // 4096 waves
    hll_partial_topk_kernel<<<grid, 32, 0, stream>>>(audio, text, norms,
                                                     ws_z, ws_i);

    hll_merge_loss_kernel<<<(2 * B_N + 255) / 256, 256, 0, stream>>>(
        audio, text, norms, ws_z, ws_i, rowloss);

    hll_reduce_kernel<<<1, 256, 0, stream>>>(rowloss, (float*)d_out);
}